// Convolution_15161234555217
// MI455X (gfx1250) — compile-verified
//
#include <hip/hip_runtime.h>

#define A_DIM 512
#define RBF_N 64
#define F_DIM 64
#define U_DIM 64
#define EPS_F 1e-7f
#define SW_STRIDE 72   // LDS row stride for W12: 72 mod 64 == 8 -> rows k, k+2 hit disjoint bank windows

typedef float v2f __attribute__((ext_vector_type(2)));
typedef float v8f __attribute__((ext_vector_type(8)));

struct F3 { float x, y, z; };   // 12-byte packet -> global_load_b96

// ---------------------------------------------------------------------------
// Kernel 0: fold the affine chain.  W12 = W1 @ W2 (64x64), beff = b1@W2 + b2.
// ws[0..4095] = W12 (rbf-major, f-minor), ws[4096..4159] = beff.
// ---------------------------------------------------------------------------
__global__ void prep_kernel(const float* __restrict__ W1, const float* __restrict__ b1,
                            const float* __restrict__ W2, const float* __restrict__ b2,
                            float* __restrict__ ws) {
  const int tid = threadIdx.x;
  for (int idx = tid; idx < RBF_N * F_DIM; idx += 256) {
    const int r = idx >> 6;
    const int f = idx & 63;
    float s = 0.f;
    for (int k = 0; k < F_DIM; ++k) s += W1[r * F_DIM + k] * W2[k * F_DIM + f];
    ws[idx] = s;
  }
  if (tid < F_DIM) {
    float s = b2[tid];
    for (int k = 0; k < F_DIM; ++k) s += b1[k] * W2[k * F_DIM + tid];
    ws[RBF_N * F_DIM + tid] = s;
  }
}

// ---------------------------------------------------------------------------
// Kernel 1: one workgroup per atom a (512 blocks, 256 threads = 8 waves).
// Wave w: b rows [128*(w>>1), +128), f-half = (w&1) -> two 16-wide f-tiles.
// Per 16-row b-tile: one clause of 16 A-fragment loads, then 32 WMMAs
// (2 independent chains), then the fused o1..o5 b-contraction epilogue.
// ---------------------------------------------------------------------------
__global__ __launch_bounds__(256) void conv_fused_kernel(
    const float* __restrict__ image, const float* __restrict__ vectors,
    const float* __restrict__ t0, const float* __restrict__ t1,
    const float* __restrict__ w_si0, const float* __restrict__ w_si1,
    const float* __restrict__ b_act0, const float* __restrict__ b_act1,
    const float* __restrict__ ws, float* __restrict__ out) {
  __shared__ float sW[RBF_N * SW_STRIDE];  // 18 KB: W12, k-major, padded rows
  __shared__ float sBe[F_DIM];
  __shared__ float sAcc[F_DIM][12];        // 11 components + pad

  const int a    = blockIdx.x;
  const int tid  = threadIdx.x;
  const int lane = tid & 31;
  const int wave = tid >> 5;

  for (int idx = tid; idx < RBF_N * F_DIM; idx += 256) {
    sW[(idx >> 6) * SW_STRIDE + (idx & 63)] = ws[idx];
  }
  if (tid < F_DIM) sBe[tid] = ws[RBF_N * F_DIM + tid];
  for (int i = tid; i < F_DIM * 12; i += 256) (&sAcc[0][0])[i] = 0.f;
  __syncthreads();

  const int lh     = lane >> 4;          // lane half 0/1
  const int ln     = lane & 15;          // 0..15
  const int fh     = wave & 1;           // f half: tiles {32*fh, 32*fh+16}
  const int b_base = (wave >> 1) * 128;  // 4 disjoint b ranges

  const int f0 = fh * 32 + ln;           // lane's f in tile 0
  const int f1 = f0 + 16;                // lane's f in tile 1

  // per-lane partials for the two f-tiles
  float p1[2], p4[2];
  float p2x[2], p2y[2], p2z[2];
  float p3x[2], p3y[2], p3z[2];
  float p5x[2], p5y[2], p5z[2];
#pragma unroll
  for (int ft = 0; ft < 2; ++ft) {
    p1[ft] = p4[ft] = 0.f;
    p2x[ft] = p2y[ft] = p2z[ft] = 0.f;
    p3x[ft] = p3y[ft] = p3z[ft] = 0.f;
    p5x[ft] = p5y[ft] = p5z[ft] = 0.f;
  }

  const float* img_a = image + (size_t)a * (A_DIM * RBF_N);
  const float* vec_a = vectors + (size_t)a * (A_DIM * 3);
  const float  be0 = sBe[f0];
  const float  be1 = sBe[f1];

  for (int bt = 0; bt < 8; ++bt) {
    const int tile_b = b_base + bt * 16;
    // prefetch next tile's A rows into L0 while this tile computes
    if (bt < 7) {
      __builtin_prefetch(img_a + (size_t)(tile_b + 16 + ln) * RBF_N + lh * 128, 0, 1);
    }

    // ---- Stage 1: one clause of 16 independent A-fragment loads ----
    // A fragment: 16x4 f32, lane holds A[m][2*lh], A[m][2*lh+1]; m = ln
    const float* arow = img_a + (size_t)(tile_b + ln) * RBF_N + 2 * lh;
    v2f afr[16];
#pragma unroll
    for (int ks = 0; ks < 16; ++ks) {
      afr[ks] = *(const v2f*)(arow + ks * 4);
    }

    // ---- Stage 2: 32 WMMAs, two independent accumulation chains ----
    v8f acc0 = {0.f,0.f,0.f,0.f,0.f,0.f,0.f,0.f};
    v8f acc1 = acc0;
#pragma unroll
    for (int ks = 0; ks < 16; ++ks) {
      // B fragment: 4x16, VGPR v holds row k = v + 2*lh, col = ln
      // (loop-invariant across bt; compiler hoists these LDS reads)
      const int krow = (ks * 4 + 2 * lh) * SW_STRIDE;
      {
        const float* bp = &sW[krow + f0];
        v2f bf; bf.x = bp[0]; bf.y = bp[SW_STRIDE];
        acc0 = __builtin_amdgcn_wmma_f32_16x16x4_f32(false, afr[ks], false, bf, (short)0, acc0, false, false);
      }
      {
        const float* bp = &sW[krow + f1];
        v2f bf; bf.x = bp[0]; bf.y = bp[SW_STRIDE];
        acc1 = __builtin_amdgcn_wmma_f32_16x16x4_f32(false, afr[ks], false, bf, (short)0, acc1, false, false);
      }
    }

    // ---- Stage 3: fold b-contractions. acc{0,1}[i] = R[a, tile_b+i+8*lh, f{0,1}] ----
#pragma unroll
    for (int i = 0; i < 8; ++i) {
      const int brow = tile_b + i + 8 * lh;
      const F3  v3   = *(const F3*)(vec_a + brow * 3);
      const float vx = v3.x, vy = v3.y, vz = v3.z;
      const float vn2  = vx * vx + vy * vy + vz * vz;
      const bool  keep = (vn2 >= EPS_F * EPS_F);  // kills only the diagonal
#pragma unroll
      for (int ft = 0; ft < 2; ++ft) {
        const int   f    = (ft == 0) ? f0 : f1;
        const float rawR = (ft == 0) ? acc0[i] : acc1[i];
        const float Rv = rawR + ((ft == 0) ? be0 : be1);
        const float Rm = keep ? Rv : 0.f;
        const float t0v = t0[brow * F_DIM + f];
        const F3    t1v = *(const F3*)(t1 + (size_t)(brow * F_DIM + f) * 3);
        p1[ft] += Rv * t0v;
        const float rt0 = Rm * t0v;
        p2x[ft] += vx * rt0; p2y[ft] += vy * rt0; p2z[ft] += vz * rt0;
        p3x[ft] += Rv * t1v.x; p3y[ft] += Rv * t1v.y; p3z[ft] += Rv * t1v.z;
        p4[ft]  += Rm * (vx * t1v.x + vy * t1v.y + vz * t1v.z);
        p5x[ft] += Rm * (vy * t1v.z - vz * t1v.y);
        p5y[ft] += Rm * (vz * t1v.x - vx * t1v.z);
        p5z[ft] += Rm * (vx * t1v.y - vy * t1v.x);
      }
    }
  }

  // Cross-wave / cross-lane-half reduction via LDS float atomics.
#pragma unroll
  for (int ft = 0; ft < 2; ++ft) {
    const int f = (ft == 0) ? f0 : f1;
    atomicAdd(&sAcc[f][0], p1[ft]);
    atomicAdd(&sAcc[f][1], p2x[ft]); atomicAdd(&sAcc[f][2], p2y[ft]); atomicAdd(&sAcc[f][3], p2z[ft]);
    atomicAdd(&sAcc[f][4], p3x[ft]); atomicAdd(&sAcc[f][5], p3y[ft]); atomicAdd(&sAcc[f][6], p3z[ft]);
    atomicAdd(&sAcc[f][7], p4[ft]);
    atomicAdd(&sAcc[f][8], p5x[ft]); atomicAdd(&sAcc[f][9], p5y[ft]); atomicAdd(&sAcc[f][10], p5z[ft]);
  }
  __syncthreads();

  // Phase C: s0 = c0 @ w_si0[g], s1 = c1 @ w_si1[g]; tiny, 64 threads suffice.
  if (tid < U_DIM) {
    const int g = tid;
    const float* w0 = w_si0 + (size_t)g * (2 * F_DIM);
    const float* w1 = w_si1 + (size_t)g * (3 * F_DIM);
    float s0 = 0.f, s1x = 0.f, s1y = 0.f, s1z = 0.f;
    for (int ff = 0; ff < F_DIM; ++ff) {
      s0 += sAcc[ff][0] * w0[ff] + sAcc[ff][7] * w0[F_DIM + ff];
      const float a2 = w1[ff];
      const float a3 = w1[F_DIM + ff];
      const float a5 = w1[2 * F_DIM + ff];
      s1x += sAcc[ff][1] * a2 + sAcc[ff][4] * a3 + sAcc[ff][8]  * a5;
      s1y += sAcc[ff][2] * a2 + sAcc[ff][5] * a3 + sAcc[ff][9]  * a5;
      s1z += sAcc[ff][3] * a2 + sAcc[ff][6] * a3 + sAcc[ff][10] * a5;
    }
    float out0v = s0 + b_act0[g];
    out0v = out0v > 0.f ? out0v : 0.f;
    float n2 = s1x * s1x + s1y * s1y + s1z * s1z;
    n2 = n2 > EPS_F ? n2 : EPS_F;
    const float n1 = sqrtf(n2);
    float a1 = n1 + b_act1[g];
    a1 = a1 > 0.f ? a1 : 0.f;
    const float scale = a1 / n1;
    out[(size_t)a * U_DIM + g] = out0v;
    float* o1p = out + (size_t)A_DIM * U_DIM + ((size_t)a * U_DIM + g) * 3;
    o1p[0] = s1x * scale;
    o1p[1] = s1y * scale;
    o1p[2] = s1z * scale;
  }
}

extern "C" void kernel_launch(void* const* d_in, const int* in_sizes, int n_in,
                              void* d_out, int out_size, void* d_ws, size_t ws_size,
                              hipStream_t stream) {
  const float* image   = (const float*)d_in[0];
  const float* vectors = (const float*)d_in[1];
  const float* t0      = (const float*)d_in[2];
  const float* t1      = (const float*)d_in[3];
  const float* W1      = (const float*)d_in[4];
  const float* b1      = (const float*)d_in[5];
  const float* W2      = (const float*)d_in[6];
  const float* b2      = (const float*)d_in[7];
  const float* w_si0   = (const float*)d_in[8];
  const float* w_si1   = (const float*)d_in[9];
  const float* b_act0  = (const float*)d_in[10];
  const float* b_act1  = (const float*)d_in[11];
  float* out = (float*)d_out;
  float* ws  = (float*)d_ws;

  prep_kernel<<<1, 256, 0, stream>>>(W1, b1, W2, b2, ws);
  conv_fused_kernel<<<A_DIM, 256, 0, stream>>>(image, vectors, t0, t1,
                                               w_si0, w_si1, b_act0, b_act1,
                                               ws, out);
}